// ISTA_4140348473785
// MI455X (gfx1250) — compile-verified
//
#include <hip/hip_runtime.h>

typedef __attribute__((ext_vector_type(2))) float v2f;
typedef __attribute__((ext_vector_type(8))) float v8f;

#define N_ROWS 2048
#define M_COLS 8192
#define MU_C   0.1f
#define RHO_C  0.01f
#define BETA_C (MU_C * RHO_C)
#define EPS_C  0.001f
#define T_ITER 100

// ---------------------------------------------------------------------------
// init: s = 0, accumulators = 0, done = 0, block-counter = 0
// ---------------------------------------------------------------------------
__global__ void ista_init(float* __restrict__ s, float* __restrict__ sqAcc,
                          float* __restrict__ l1Acc, int* __restrict__ done,
                          int* __restrict__ counter) {
    const int gid = blockIdx.x * blockDim.x + threadIdx.x;
    if (gid < M_COLS) s[gid] = 0.0f;
    if (gid <= T_ITER) { sqAcc[gid] = 0.0f; l1Acc[gid] = 0.0f; }
    if (gid == 0) { *done = 0; *counter = 0; }
}

// ---------------------------------------------------------------------------
// Row GEMV: temp = H @ s - x ; sqPartial[block] = sum(temp^2) over 16 rows.
// One block = 16 rows, 8 waves split K (M) range. WMMA f32 16x16x4:
//   A (16x4 of H): lanes 0-15 -> row l, K{0,1} (b64); lanes 16-31 -> row l-16, K{2,3}
//   B (4x16): s[k..k+3] replicated in every column == per-lane b64 of s with
//             byte offset (lane<16 ? 0 : 8). All D columns == row sums.
// When done is set, s/temp/sqPartial are frozen: skip the sweep entirely.
// ---------------------------------------------------------------------------
__global__ void ista_row_gemv(const float* __restrict__ H, const float* __restrict__ x,
                              const float* __restrict__ s, float* __restrict__ temp,
                              float* __restrict__ sqPartial, const int* __restrict__ done) {
    if (*done != 0) return;                      // uniform: converged, everything frozen
    __shared__ float red[8][16];
    __shared__ float sq[16];
    const int tid   = threadIdx.x;
    const int wave  = tid >> 5;
    const int lane  = tid & 31;
    const int row0  = blockIdx.x * 16;
    const int rlane = lane & 15;
    const int koff  = (lane >> 4) << 1;          // 0 (lanes 0-15) or 2 (lanes 16-31)
    const int kb0   = wave * (M_COLS / 8);       // 1024 K per wave

    const float* ap = H + (size_t)(row0 + rlane) * M_COLS + kb0 + koff;
    const float* bp = s + kb0 + koff;

    v8f acc = {};
    for (int k = 0; k < M_COLS / 8; k += 4) {
        v2f a = *(const v2f*)ap;                 // H[row][kb+koff .. +1]
        v2f b = *(const v2f*)bp;                 // s[kb+koff .. +1] (replicated-B trick)
        ap += 4; bp += 4;
        acc = __builtin_amdgcn_wmma_f32_16x16x4_f32(false, a, false, b,
                                                    (short)0, acc, false, false);
    }
    // D layout: lane 0, VGPR v -> row v ; lane 16, VGPR v -> row 8+v (col 0)
    if (lane == 0) {
        for (int v = 0; v < 8; ++v) red[wave][v] = acc[v];
    } else if (lane == 16) {
        for (int v = 0; v < 8; ++v) red[wave][8 + v] = acc[v];
    }
    __syncthreads();
    if (tid < 16) {
        float t = 0.0f;
        for (int w = 0; w < 8; ++w) t += red[w][tid];
        t -= x[row0 + tid];
        temp[row0 + tid] = t;
        sq[tid] = t * t;
    }
    __syncthreads();
    if (tid == 0) {
        float q = 0.0f;
        for (int i = 0; i < 16; ++i) q += sq[i];
        sqPartial[blockIdx.x] = q;
    }
}

// ---------------------------------------------------------------------------
// Col GEMV + shrinkage: s_new = shrink(s - MU * H^T @ temp, BETA) unless done.
// One wave owns 16 columns (8 waves/block, 64 blocks). WMMA f32 16x16x4:
//   A (16x4): temp[n..n+3] replicated in every row == per-lane b64 of temp
//   B (4x16 of H): lane j loads H[n+{0,1 or 2,3}][m0+j] (64B row segments)
// All D rows equal the column dot products; lanes 0-15 hold column m0+lane.
// The LAST finishing block (fence + atomic counter) performs the per-iteration
// scalar bookkeeping in fixed order (deterministic):
//   sqAcc[k]   = sum sqPartial   (= ||H@s_k - x||^2)
//   l1Acc[k+1] = sum l1Partial   (= ||s_{k+1}||_1)
//   done      |= (sum deltaPartial <= EPS)
// ---------------------------------------------------------------------------
__global__ void ista_col_gemv(const float* __restrict__ H, const float* __restrict__ temp,
                              float* __restrict__ s, int* __restrict__ done,
                              float* __restrict__ deltaPartial, float* __restrict__ l1Partial,
                              const float* __restrict__ sqPartial,
                              float* __restrict__ sqAcc, float* __restrict__ l1Acc,
                              int* __restrict__ counter, int k) {
    __shared__ float redD[256];
    __shared__ float redL[256];
    const int tid  = threadIdx.x;
    const int wave = tid >> 5;
    const int lane = tid & 31;
    const int m0   = (blockIdx.x * 8 + wave) * 16;
    const int jl   = lane & 15;
    const int nk   = (lane >> 4) << 1;           // 0 or 2

    float dsum = 0.0f, lsum = 0.0f;
    const bool isDone = (*done != 0);
    if (!isDone) {
        const float* tp  = temp + nk;
        const float* bp0 = H + (size_t)nk * M_COLS + m0 + jl;
        const float* bp1 = bp0 + M_COLS;
        v8f acc = {};
        for (int n = 0; n < N_ROWS; n += 4) {
            v2f a = *(const v2f*)tp; tp += 4;    // temp[n+nk], temp[n+nk+1]
            v2f b; b.x = *bp0; b.y = *bp1;       // H[n+nk][m], H[n+nk+1][m]
            bp0 += 4 * (size_t)M_COLS; bp1 += 4 * (size_t)M_COLS;
            acc = __builtin_amdgcn_wmma_f32_16x16x4_f32(false, a, false, b,
                                                        (short)0, acc, false, false);
        }
        if (lane < 16) {
            const int   m   = m0 + jl;
            const float sp  = s[m];
            const float g   = sp - MU_C * acc[0];
            const float mag = fabsf(g) - BETA_C;
            const float sn  = (mag > 0.0f) ? copysignf(mag, g) : 0.0f;
            dsum = fabsf(sn - sp);
            lsum = fabsf(sn);
            s[m] = sn;
        }
    } else {
        if (lane < 16) lsum = fabsf(s[m0 + jl]); // frozen iterate: l1 only
    }
    redD[tid] = dsum; redL[tid] = lsum;
    __syncthreads();
    for (int st = 128; st > 0; st >>= 1) {
        if (tid < st) { redD[tid] += redD[tid + st]; redL[tid] += redL[tid + st]; }
        __syncthreads();
    }
    if (tid == 0) {
        deltaPartial[blockIdx.x] = redD[0];
        l1Partial[blockIdx.x]    = redL[0];
        __threadfence();
        if (atomicAdd(counter, 1) == (int)gridDim.x - 1) {
            __threadfence();                     // acquire all blocks' partials
            float q = 0.0f;
            for (int i = 0; i < N_ROWS / 16; ++i) q += sqPartial[i];
            sqAcc[k] = q;
            float d = 0.0f, l = 0.0f;
            for (int i = 0; i < M_COLS / 128; ++i) { d += deltaPartial[i]; l += l1Partial[i]; }
            l1Acc[k + 1] = l;
            if (d <= EPS_C) *done = 1;
            *counter = 0;                        // re-arm for next launch / replay
        }
    }
}

// ---------------------------------------------------------------------------
// Finalize: out[0..8191] = s ; out[8192+j] = 0.5*sqAcc[j+1] + RHO*l1Acc[j+1]
// (sqAcc[100] comes from the extra row pass, summed here)
// ---------------------------------------------------------------------------
__global__ void ista_final(const float* __restrict__ s, const float* __restrict__ sqPartial,
                           const float* __restrict__ sqAcc, const float* __restrict__ l1Acc,
                           float* __restrict__ out) {
    const int gid = blockIdx.x * blockDim.x + threadIdx.x;
    if (gid < M_COLS) out[gid] = s[gid];
    if (blockIdx.x == gridDim.x - 1) {
        __shared__ float lastSq;
        if (threadIdx.x == 0) {
            float q = 0.0f;
            for (int i = 0; i < N_ROWS / 16; ++i) q += sqPartial[i];
            lastSq = q;
        }
        __syncthreads();
        if (threadIdx.x < T_ITER) {
            const int   j  = threadIdx.x;
            const float sq = (j == T_ITER - 1) ? lastSq : sqAcc[j + 1];
            out[M_COLS + j] = 0.5f * sq + RHO_C * l1Acc[j + 1];
        }
    }
}

// ---------------------------------------------------------------------------
extern "C" void kernel_launch(void* const* d_in, const int* in_sizes, int n_in,
                              void* d_out, int out_size, void* d_ws, size_t ws_size,
                              hipStream_t stream) {
    (void)in_sizes; (void)n_in; (void)out_size; (void)ws_size;
    const float* H = (const float*)d_in[0];
    const float* x = (const float*)d_in[1];
    // d_in[2] = max_iter (fixed at 100; cannot be read on host inside capture)

    float* ws      = (float*)d_ws;
    float* s       = ws;                    // 8192
    float* temp    = ws + 8192;             // 2048
    float* sqP     = temp + 2048;           // 128
    float* dP      = sqP + 128;             // 64
    float* lP      = dP + 64;               // 64
    float* sqAcc   = lP + 64;               // 101 (padded 104)
    float* l1Acc   = sqAcc + 104;           // 101 (padded 104)
    int*   done    = (int*)(l1Acc + 104);   // 1
    int*   counter = done + 1;              // 1

    ista_init<<<32, 256, 0, stream>>>(s, sqAcc, l1Acc, done, counter);
    for (int k = 0; k < T_ITER; ++k) {
        ista_row_gemv<<<N_ROWS / 16, 256, 0, stream>>>(H, x, s, temp, sqP, done);
        ista_col_gemv<<<M_COLS / 128, 256, 0, stream>>>(H, temp, s, done, dP, lP,
                                                        sqP, sqAcc, l1Acc, counter, k);
    }
    ista_row_gemv<<<N_ROWS / 16, 256, 0, stream>>>(H, x, s, temp, sqP, done); // err[99]
    ista_final<<<33, 256, 0, stream>>>(s, sqP, sqAcc, l1Acc, (float*)d_out);
}